// Loss_16544214024588
// MI455X (gfx1250) — compile-verified
//
#include <hip/hip_runtime.h>
#include <hip/hip_bf16.h>
#include <math.h>

typedef __attribute__((ext_vector_type(2))) float v2f;
typedef __attribute__((ext_vector_type(8))) float v8f;
typedef __attribute__((ext_vector_type(4))) unsigned int u32x4;
typedef __attribute__((ext_vector_type(4))) int          i32x4;
typedef __attribute__((ext_vector_type(8))) int          i32x8;

#define BDIM 8192
#define DDIM 128
#define TM   128          // block tile M (anchor rows)
#define TN   128          // block tile N (positive rows)
#define LDS_STRIDE 132    // 128 + 4-DWORD TDM pad -> conflict-free frag reads

// ---------------------------------------------------------------------------
// Kernel 1: per-row inverse norm  inv[r] = 1 / max(||X[r,:]||, 1e-8)
// ---------------------------------------------------------------------------
__global__ void row_invnorm_kernel(const float* __restrict__ X,
                                   float* __restrict__ inv, int nrows) {
    int gtid = blockIdx.x * blockDim.x + threadIdx.x;
    int row  = gtid >> 5;
    int lane = threadIdx.x & 31;
    if (row >= nrows) return;
    const float4* r = (const float4*)(X + (size_t)row * DDIM);
    float4 v = r[lane];
    float s = v.x * v.x + v.y * v.y + v.z * v.z + v.w * v.w;
#pragma unroll
    for (int off = 16; off > 0; off >>= 1)
        s += __shfl_xor(s, off, 32);
    if (lane == 0)
        inv[row] = 1.0f / fmaxf(sqrtf(s), 1e-8f);
}

// ---------------------------------------------------------------------------
// TDM: load a 128-row x 128-float (512B) 2D tile into LDS.
// D# pad: after every 128 DWORDs stored, skip 4 DWORDs -> LDS row stride 132.
//   group1.w0: data_size=2 (4B) | pad_enable | pad_interval=6 (128 DW)
//              | pad_amount=3 (4 DW)
// ---------------------------------------------------------------------------
__device__ __forceinline__ void tdm_load_tile(const float* gptr,
                                              unsigned ldsOff, int rows) {
    unsigned long long ga = (unsigned long long)(uintptr_t)gptr;
    u32x4 g0;
    g0[0] = 1u;                                    // count=1, user descriptor
    g0[1] = ldsOff;                                // lds_addr (bytes)
    g0[2] = (unsigned)(ga & 0xFFFFFFFFu);          // global_addr[31:0]
    g0[3] = ((unsigned)(ga >> 32) & 0x01FFFFFFu)   // global_addr[56:32]
            | 0x80000000u;                         // type = 2 ("image")
    i32x8 g1;
    g1[0] = (2 << 16) | (1 << 20) | (6 << 22) | (3 << 25);
    g1[1] = (int)(128u << 16);                     // tensor_dim0 = 128
    g1[2] = (int)(8192u << 16);                    // tensor_dim1 = 8192
    g1[3] = (int)(128u << 16);                     // tile_dim0 = 128
    g1[4] = rows;                                  // tile_dim1 ; tile_dim2 = 0
    g1[5] = 128;                                   // tensor_dim0_stride = 128
    g1[6] = 0;
    g1[7] = 0;
    i32x4 z4 = {0, 0, 0, 0};                       // groups 2/3: 2D tensor
#if __clang_major__ >= 23
    i32x8 z8 = {0, 0, 0, 0, 0, 0, 0, 0};
    __builtin_amdgcn_tensor_load_to_lds(g0, g1, z4, z4, z8, 0);
#else
    __builtin_amdgcn_tensor_load_to_lds(g0, g1, z4, z4, 0);
#endif
}

// ---------------------------------------------------------------------------
// Kernel 2: 128x128 block tile, 8 waves, wave tile 32x64 (2x4 subtiles of
// 16x16). Raw fp32 WMMA GEMM + fused cosine-loss epilogue.
// ---------------------------------------------------------------------------
__global__ void cos_loss_gemm_kernel(const float* __restrict__ A,   // anchor  [B,D]
                                     const float* __restrict__ P,   // positive[B,D]
                                     const float* __restrict__ inva,
                                     const float* __restrict__ invp,
                                     float* __restrict__ partials) {
    __shared__ float smem[2 * TM * LDS_STRIDE];    // ~132 KB of 320 KB LDS
    float* As = smem;
    float* Ps = smem + TM * LDS_STRIDE;

    const int tid  = threadIdx.x;
    const int bx   = blockIdx.x;    // N tile (positive rows)
    const int by   = blockIdx.y;    // M tile (anchor rows)
    const int lane = tid & 31;
    const int wav  = tid >> 5;

    // ---- TDM fill: wave 0 -> A tile, wave 1 -> P tile
    if (wav == 0) {
        tdm_load_tile(A + (size_t)(by * TM) * DDIM,
                      (unsigned)(uintptr_t)As, TM);
    } else if (wav == 1) {
        tdm_load_tile(P + (size_t)(bx * TN) * DDIM,
                      (unsigned)(uintptr_t)Ps, TN);
    }
    if (wav < 2) __builtin_amdgcn_s_wait_tensorcnt(0);
    __syncthreads();

    // ---- wave tile: wm in 0..3 (rows), wn in 0..1 (cols)
    const int wm   = wav & 3;                  // 32-row group
    const int wn   = wav >> 2;                 // 64-col group
    const int frow = lane & 15;                // M/N within a 16-subtile
    const int kofs = (lane >> 4) << 1;         // upper half-lanes: K+2,K+3

    const float* aB0 = &As[(wm * 32 +  0 + frow) * LDS_STRIDE + kofs];
    const float* aB1 = &As[(wm * 32 + 16 + frow) * LDS_STRIDE + kofs];
    const float* bB0 = &Ps[(wn * 64 +  0 + frow) * LDS_STRIDE + kofs];
    const float* bB1 = &Ps[(wn * 64 + 16 + frow) * LDS_STRIDE + kofs];
    const float* bB2 = &Ps[(wn * 64 + 32 + frow) * LDS_STRIDE + kofs];
    const float* bB3 = &Ps[(wn * 64 + 48 + frow) * LDS_STRIDE + kofs];

    v8f acc[2][4] = {};
#pragma unroll
    for (int k = 0; k < DDIM; k += 4) {
        v2f a0 = *(const v2f*)(aB0 + k);
        v2f a1 = *(const v2f*)(aB1 + k);
        v2f b0 = *(const v2f*)(bB0 + k);
        v2f b1 = *(const v2f*)(bB1 + k);
        v2f b2 = *(const v2f*)(bB2 + k);
        v2f b3 = *(const v2f*)(bB3 + k);
        acc[0][0] = __builtin_amdgcn_wmma_f32_16x16x4_f32(false, a0, false, b0, (short)0, acc[0][0], false, false);
        acc[0][1] = __builtin_amdgcn_wmma_f32_16x16x4_f32(false, a0, false, b1, (short)0, acc[0][1], false, false);
        acc[0][2] = __builtin_amdgcn_wmma_f32_16x16x4_f32(false, a0, false, b2, (short)0, acc[0][2], false, false);
        acc[0][3] = __builtin_amdgcn_wmma_f32_16x16x4_f32(false, a0, false, b3, (short)0, acc[0][3], false, false);
        acc[1][0] = __builtin_amdgcn_wmma_f32_16x16x4_f32(false, a1, false, b0, (short)0, acc[1][0], false, false);
        acc[1][1] = __builtin_amdgcn_wmma_f32_16x16x4_f32(false, a1, false, b1, (short)0, acc[1][1], false, false);
        acc[1][2] = __builtin_amdgcn_wmma_f32_16x16x4_f32(false, a1, false, b2, (short)0, acc[1][2], false, false);
        acc[1][3] = __builtin_amdgcn_wmma_f32_16x16x4_f32(false, a1, false, b3, (short)0, acc[1][3], false, false);
    }

    // ---- epilogue: cos scale + diagonal / ReLU mask, per-lane sum
    // C layout: lanes 0-15 -> M = v, lanes 16-31 -> M = v + 8; N = lane & 15.
    const int lh = lane >> 4;
    float local = 0.0f;
#pragma unroll
    for (int s = 0; s < 2; ++s) {
        float ia[8];
        int   ib[8];
#pragma unroll
        for (int v = 0; v < 8; ++v) {
            ib[v] = by * TM + wm * 32 + s * 16 + v + 8 * lh;
            ia[v] = inva[ib[v]];
        }
#pragma unroll
        for (int t = 0; t < 4; ++t) {
            int   n  = bx * TN + wn * 64 + t * 16 + frow;
            float ip = invp[n];
#pragma unroll
            for (int v = 0; v < 8; ++v) {
                float c = acc[s][t][v] * ia[v] * ip;
                local += (ib[v] == n) ? (1.0f - c) : fmaxf(c, 0.0f);
            }
        }
    }

    // ---- block reduction (reuse LDS after barrier)
    __syncthreads();
    smem[tid] = local;
    __syncthreads();
#pragma unroll
    for (int off = 128; off > 0; off >>= 1) {
        if (tid < off) smem[tid] += smem[tid + off];
        __syncthreads();
    }
    if (tid == 0)
        partials[blockIdx.y * gridDim.x + blockIdx.x] = smem[0];
}

// ---------------------------------------------------------------------------
// Kernel 3: final reduction of block partials -> mean loss scalar
// ---------------------------------------------------------------------------
__global__ void finalize_kernel(const float* __restrict__ partials, int n,
                                float* __restrict__ out) {
    __shared__ float sh[256];
    float s = 0.0f;
    for (int i = threadIdx.x; i < n; i += 256) s += partials[i];
    sh[threadIdx.x] = s;
    __syncthreads();
#pragma unroll
    for (int off = 128; off > 0; off >>= 1) {
        if (threadIdx.x < off) sh[threadIdx.x] += sh[threadIdx.x + off];
        __syncthreads();
    }
    if (threadIdx.x == 0)
        out[0] = sh[0] * (1.0f / ((float)BDIM * (float)BDIM));
}

// ---------------------------------------------------------------------------
extern "C" void kernel_launch(void* const* d_in, const int* in_sizes, int n_in,
                              void* d_out, int out_size, void* d_ws, size_t ws_size,
                              hipStream_t stream) {
    const float* hid_positive = (const float*)d_in[0];
    const float* hid_anchor   = (const float*)d_in[1];
    float* out = (float*)d_out;

    // workspace (floats): inva[8192] | invp[8192] | partials[4096]
    float* inva     = (float*)d_ws;
    float* invp     = inva + BDIM;
    float* partials = invp + BDIM;
    const int nPartials = (BDIM / TM) * (BDIM / TN);   // 4096

    {
        dim3 block(256);
        dim3 grid((BDIM * 32 + 255) / 256);
        row_invnorm_kernel<<<grid, block, 0, stream>>>(hid_anchor,   inva, BDIM);
        row_invnorm_kernel<<<grid, block, 0, stream>>>(hid_positive, invp, BDIM);
    }
    {
        dim3 block(256);
        dim3 grid(BDIM / TN, BDIM / TM);               // 64 x 64
        cos_loss_gemm_kernel<<<grid, block, 0, stream>>>(
            hid_anchor, hid_positive, inva, invp, partials);
    }
    finalize_kernel<<<1, 256, 0, stream>>>(partials, nPartials, out);
}